// Sampler_11819749999162
// MI455X (gfx1250) — compile-verified
//
#include <hip/hip_runtime.h>
#include <stdint.h>

// Problem constants (from reference: B=128, V=128000)
#define BN    128
#define VN    128000
#define TILE  4000          // floats per TDM tile (16 KB); 32 tiles per row
#define NTILE (VN / TILE)
#define KTOP  2048          // top_k <= 2048 => filtered has <=2048 nonzeros
#define NBINS 4096          // 12-bit selection histogram
#define CAND  4096          // candidate buffer (power of two for bitonic)
#define NT    1024          // threads per block = 32 waves (wave32)

typedef __attribute__((ext_vector_type(4))) unsigned int v4u;
typedef __attribute__((ext_vector_type(8))) int          v8i;
typedef __attribute__((ext_vector_type(4))) int          v4i;
typedef __attribute__((ext_vector_type(2))) float        v2f;
typedef __attribute__((ext_vector_type(8))) float        v8f;

// ---------------------------------------------------------------------------
// Tensor Data Mover: stage one contiguous TILE*4-byte span from global into
// LDS. Descriptor built per CDNA5 ISA (2D tensor,
// data_size=4B, tile_dim0=TILE, tile_dim1=1). Issued by one wave (EXEC
// ignored by TDM); completion tracked with TENSORcnt.
// This toolchain (clang-23 / therock HIP headers) uses the 6-arg builtin:
//   (uint32x4 g0, int32x8 g1, int32x4 g2, int32x4 g3, int32x8 g4, i32 cpol)
// ---------------------------------------------------------------------------
__device__ __forceinline__ void tdm_load_tile(const float* gsrc, void* lds_dst) {
#if __has_builtin(__builtin_amdgcn_tensor_load_to_lds)
  unsigned lds = (unsigned)(uintptr_t)lds_dst;   // flat LDS addr low 32 bits = LDS offset
  unsigned long long ga = (unsigned long long)(uintptr_t)gsrc;
  v4u g0;
  g0.x = 1u;                                             // count=1, user mode, no gather
  g0.y = lds;                                            // lds_addr [63:32]
  g0.z = (unsigned)ga;                                   // global_addr lo
  g0.w = (unsigned)((ga >> 32) & 0x01FFFFFFull)          // global_addr[56:32]
         | 0x80000000u;                                  // type=2 ("image") at [127:126]
  v8i g1;
  g1[0] = 0x00020000;                                    // data_size=2 (4 bytes)
  g1[1] = (int)((TILE & 0xFFFF) << 16);                  // tensor_dim0 lo16 at [63:48]
  g1[2] = (int)(((TILE >> 16) & 0xFFFF) | (1 << 16));    // tensor_dim0 hi16, tensor_dim1=1 lo16
  g1[3] = (int)((TILE & 0xFFFF) << 16);                  // tensor_dim1 hi=0, tile_dim0=TILE
  g1[4] = 1;                                             // tile_dim1=1, tile_dim2=0
  g1[5] = TILE;                                          // tensor_dim0_stride lo32
  g1[6] = 0;
  g1[7] = 0;
  v4i gz = {0, 0, 0, 0};
  v8i gz8 = {0, 0, 0, 0, 0, 0, 0, 0};
  __builtin_amdgcn_tensor_load_to_lds(g0, g1, gz, gz, gz8, 0);
#else
  float* d = (float*)lds_dst;
  for (int i = (int)(threadIdx.x & 31); i < TILE; i += 32) d[i] = gsrc[i];
#endif
}

__device__ __forceinline__ void tdm_wait() {
#if __has_builtin(__builtin_amdgcn_s_wait_tensorcnt)
  __builtin_amdgcn_s_wait_tensorcnt(0);
#endif
}

// Order-preserving key: descending float order == descending unsigned key order
__device__ __forceinline__ unsigned fkey(float x) {
  unsigned u = __float_as_uint(x);
  return (u & 0x80000000u) ? ~u : (u | 0x80000000u);
}

// ---------------------------------------------------------------------------
// WMMA-accelerated inclusive prefix sum of in[0..KTOP-1] -> out[0..KTOP-1].
// Prefix over 16 consecutive elements == D = L*X with L = 16x16 lower-
// triangular ones, built from four chained V_WMMA_F32_16X16X4_F32 (K=4).
// Wave w scans tile w of 256 elements (column c of X = elements c*16..c*16+15).
// Layouts per CDNA5 ISA:
//   A 16x4 f32 : VGPR0 = {K0 | K2}, VGPR1 = {K1 | K3} over lane halves, M=lane%16
//   B 4x16 f32 : same K split over lane halves, N=lane%16
//   C/D 16x16  : VGPR v = rows {v | v+8}, N=lane%16
// Group totals (row 15) are offset-combined afterwards.
// ---------------------------------------------------------------------------
__device__ __forceinline__ void wmma_scan_2048(float* in, float* out,
                                               float* gsum, int tid) {
#if __has_builtin(__builtin_amdgcn_wmma_f32_16x16x4_f32)
  const int wid  = tid >> 5;
  const int lane = tid & 31;
  if (wid < (KTOP / 256)) {                 // waves 0..7, EXEC all-ones per wave
    const int  base = wid * 256;
    const int  col  = lane & 15;
    const bool hi   = lane >= 16;
    const int  row  = col;                  // A-matrix row for this lane
    v8f acc = {};
#pragma unroll
    for (int q = 0; q < 4; ++q) {
      const int k0 = 4 * q + (hi ? 2 : 0);
      const int k1 = k0 + 1;
      v2f a, bm;
      a.x  = (k0 <= row) ? 1.0f : 0.0f;     // L[row][k0]
      a.y  = (k1 <= row) ? 1.0f : 0.0f;     // L[row][k1]
      bm.x = in[base + col * 16 + k0];      // X[k0][col]
      bm.y = in[base + col * 16 + k1];      // X[k1][col]
      acc = __builtin_amdgcn_wmma_f32_16x16x4_f32(false, a, false, bm,
                                                  (short)0, acc, false, false);
    }
#pragma unroll
    for (int v = 0; v < 8; ++v)             // D[r][c] -> within-group prefix
      out[base + col * 16 + v + (hi ? 8 : 0)] = acc[v];
    if (hi) gsum[wid * 16 + col] = acc[7];  // D[15][c] = group total
  }
  __syncthreads();
  if (tid == 0) {                           // exclusive scan of 128 group sums
    float run = 0.0f;
    for (int g = 0; g < KTOP / 16; ++g) { float t = gsum[g]; gsum[g] = run; run += t; }
  }
  __syncthreads();
  for (int i = tid; i < KTOP; i += NT) out[i] += gsum[i >> 4];
  __syncthreads();
#else
  // Fallback: Hillis-Steele ping-pong (11 steps -> result lands in `out`)
  float* s = in; float* d = out;
  for (int dd = 1; dd < KTOP; dd <<= 1) {
    for (int i = tid; i < KTOP; i += NT)
      d[i] = (i >= dd) ? (s[i] + s[i - dd]) : s[i];
    __syncthreads();
    float* t = s; s = d; d = t;
  }
#endif
}

// ===========================================================================
// Kernel 1: per-row max, softmax normalizer, and 12-bit selection histogram.
// Phase 2 streams the row via double-buffered TDM tiles into LDS.
// ===========================================================================
__global__ __launch_bounds__(NT)
void sampler_stats_kernel(const float* __restrict__ logits,
                          const float* __restrict__ temps,
                          float* __restrict__ sMaxOut,
                          float* __restrict__ zOut,
                          int*   __restrict__ thrOut) {
  __shared__ float    stage[2][TILE];   // 32 KB double buffer
  __shared__ unsigned hist[NBINS];      // 16 KB
  __shared__ float    red[NT / 32];
  __shared__ float    sh_max;

  const int b   = blockIdx.x;
  const int tid = (int)threadIdx.x;
  const float* row = logits + (size_t)b * VN;

  for (int i = tid; i < NBINS; i += NT) hist[i] = 0u;

  // ---- Phase 1: row max via coalesced b128 loads ----
  float m = -3.402823466e38f;
  const float4* row4 = (const float4*)row;
  for (int i = tid; i < VN / 4; i += NT) {
    float4 v = row4[i];
    m = fmaxf(m, fmaxf(fmaxf(v.x, v.y), fmaxf(v.z, v.w)));
  }
  for (int off = 16; off; off >>= 1) m = fmaxf(m, __shfl_xor(m, off));
  if ((tid & 31) == 0) red[tid >> 5] = m;
  __syncthreads();
  if (tid < 32) {
    float t = red[tid];
    for (int off = 16; off; off >>= 1) t = fmaxf(t, __shfl_xor(t, off));
    if (tid == 0) sh_max = t;
  }
  __syncthreads();

  const float invT = 1.0f / temps[b];
  const float smax = sh_max * invT;     // max of logits/T (T > 0)

  // ---- Phase 2: TDM-staged stream -> sum(exp) + histogram ----
  if (tid < 32) { tdm_load_tile(row, &stage[0][0]); tdm_wait(); }
  __syncthreads();
  float sum = 0.0f;
  for (int t = 0; t < NTILE; ++t) {
    const int cur = t & 1;
    if (tid < 32 && (t + 1) < NTILE)
      tdm_load_tile(row + (size_t)(t + 1) * TILE, &stage[cur ^ 1][0]);
    for (int i = tid; i < TILE; i += NT) {
      float x = stage[cur][i];
      sum += __expf(x * invT - smax);
      atomicAdd(&hist[fkey(x) >> 20], 1u);
    }
    __syncthreads();
    if (tid < 32) tdm_wait();           // next tile resident before anyone reads it
    __syncthreads();
  }

  for (int off = 16; off; off >>= 1) sum += __shfl_xor(sum, off);
  if ((tid & 31) == 0) red[tid >> 5] = sum;
  __syncthreads();
  if (tid == 0) {
    float z = 0.0f;
    for (int i = 0; i < NT / 32; ++i) z += red[i];
    zOut[b]    = z;
    sMaxOut[b] = smax;
    unsigned cum = 0;
    int bin = NBINS - 1;
    for (; bin > 0; --bin) {            // bin containing the KTOP-th largest
      unsigned h = hist[bin];
      if (cum + h >= (unsigned)KTOP) break;
      cum += h;
    }
    thrOut[b] = bin;
  }
}

// ===========================================================================
// Kernel 2: compact candidates >= threshold bin, bitonic sort (key desc,
// index asc = stable argsort tie-break), softmax values, WMMA scans for
// top-p mask and sampling CDF, write filtered head + zero tail, sample.
// ===========================================================================
__global__ __launch_bounds__(NT)
void sampler_select_kernel(const float* __restrict__ logits,
                           const float* __restrict__ temps,
                           const int*   __restrict__ top_ks,
                           const float* __restrict__ top_ps,
                           const float* __restrict__ min_ps,
                           const int*   __restrict__ need_minp,
                           const float* __restrict__ sMaxIn,
                           const float* __restrict__ zIn,
                           const int*   __restrict__ thrIn,
                           float* __restrict__ out_ids,
                           float* __restrict__ out_filtered) {
  __shared__ unsigned candKey[CAND];    // 16 KB
  __shared__ int      candIdx[CAND];    // 16 KB
  __shared__ float    pa[KTOP];         // 8 KB (values / filtered)
  __shared__ float    pb[KTOP];         // 8 KB (cumsum)
  __shared__ float    gsum[KTOP / 16];  // 512 B group offsets for WMMA scan
  __shared__ int      s_cnt;

  const int b   = blockIdx.x;
  const int tid = (int)threadIdx.x;
  const float* row = logits + (size_t)b * VN;
  const unsigned thr = (unsigned)thrIn[b];

  if (tid == 0) s_cnt = 0;
  __syncthreads();

  // ---- Compaction: everything with 12-bit key prefix >= thr is a candidate
  const float4* row4 = (const float4*)row;
  for (int i4 = tid; i4 < VN / 4; i4 += NT) {
    __builtin_prefetch(row4 + i4 + 2 * NT, 0, 1);   // global_prefetch_b8
    float4 v = row4[i4];
    float vv[4] = {v.x, v.y, v.z, v.w};
#pragma unroll
    for (int c = 0; c < 4; ++c) {
      unsigned key = fkey(vv[c]);
      if ((key >> 20) >= thr) {
        int p = atomicAdd(&s_cnt, 1);
        if (p < CAND) { candKey[p] = key; candIdx[p] = i4 * 4 + c; }
      }
    }
  }
  __syncthreads();
  const int cnt = min(s_cnt, CAND);
  for (int i = cnt + tid; i < CAND; i += NT) { candKey[i] = 0u; candIdx[i] = 0x7FFFFFFF; }
  __syncthreads();

  // ---- Bitonic sort of CAND pairs: key desc, index asc on ties ----
  for (int k = 2; k <= CAND; k <<= 1) {
    for (int j = k >> 1; j > 0; j >>= 1) {
      for (int t = tid; t < CAND; t += NT) {
        const int ixj = t ^ j;
        if (ixj > t) {
          unsigned ka = candKey[t];   int ia = candIdx[t];
          unsigned kb = candKey[ixj]; int ib = candIdx[ixj];
          const bool b_before_a = (kb > ka) || (kb == ka && ib < ia);
          const bool a_before_b = (ka > kb) || (ka == kb && ia < ib);
          const bool sw = ((t & k) == 0) ? b_before_a : a_before_b;
          if (sw) { candKey[t] = kb; candIdx[t] = ib; candKey[ixj] = ka; candIdx[ixj] = ia; }
        }
      }
      __syncthreads();
    }
  }

  // ---- Softmax values for the top-KTOP ----
  const float smax = sMaxIn[b];
  const float invZ = 1.0f / zIn[b];
  const float invT = 1.0f / temps[b];
  for (int i = tid; i < KTOP; i += NT) {
    unsigned key = candKey[i];
    unsigned u   = (key & 0x80000000u) ? (key & 0x7FFFFFFFu) : ~key;  // invert fkey
    float p = __expf(__uint_as_float(u) * invT - smax) * invZ;
    if (i >= cnt) p = 0.0f;
    pa[i] = p;
  }
  __syncthreads();

  // ---- WMMA inclusive scan of sorted probs: pb = cumsum(pa) ----
  wmma_scan_2048(pa, pb, gsum, tid);

  const int   topk = top_ks[b];
  const float topp = top_ps[b];
  const float minp = min_ps[b];
  const int   nmp  = need_minp[0];
  const float p0   = pb[0];                         // rank 0 is always kept
  const float thrv = nmp ? (p0 * minp) : -1.0f;
  float* orow = out_filtered + (size_t)b * VN;
  for (int i = tid; i < KTOP; i += NT) {
    float cum = pb[i];
    float p   = cum - (i ? pb[i - 1] : 0.0f);
    bool keep = (i < topk) && ((cum - p) <= topp);  // exclusive-cumsum top-p test
    float f = keep ? p : 0.0f;
    if (f < thrv) f = 0.0f;                         // min-p
    orow[i] = f;
    pa[i]   = f;
  }
  __syncthreads();

  // ---- Zero tail of filtered output (ranks >= KTOP >= top_k are all zero)
  float4 z4; z4.x = z4.y = z4.z = z4.w = 0.0f;
  float4* otail = (float4*)(orow + KTOP);
  for (int i = tid; i < (VN - KTOP) / 4; i += NT) otail[i] = z4;

  // ---- WMMA inclusive scan of filtered mass: pb = cumF ----
  wmma_scan_2048(pa, pb, gsum, tid);

  if (tid == 0) {
    const float total = pb[KTOP - 1];
    // deterministic per-row hash RNG (JAX threefry bits are not reproducible)
    unsigned st = 0x9E3779B9u * (unsigned)(b + 1) + 42u;
    st ^= st >> 16; st *= 0x7FEB352Du;
    st ^= st >> 15; st *= 0x846CA68Bu;
    st ^= st >> 16;
    const float u01    = (float)(st >> 8) * (1.0f / 16777216.0f);
    const float target = u01 * total;
    int lo = 0, hi = KTOP - 1;
    while (lo < hi) {                   // first index with cumF > target
      const int mid = (lo + hi) >> 1;
      if (pb[mid] > target) hi = mid; else lo = mid + 1;
    }
    out_ids[b] = (float)candIdx[lo];
  }
}

// ===========================================================================
extern "C" void kernel_launch(void* const* d_in, const int* in_sizes, int n_in,
                              void* d_out, int out_size, void* d_ws, size_t ws_size,
                              hipStream_t stream) {
  (void)in_sizes; (void)n_in; (void)out_size; (void)ws_size;
  const float* logits = (const float*)d_in[0];
  const float* temps  = (const float*)d_in[1];
  const int*   topks  = (const int*)d_in[2];
  const float* topps  = (const float*)d_in[3];
  const float* minps  = (const float*)d_in[4];
  const int*   nmp    = (const int*)d_in[5];

  float* out_ids      = (float*)d_out;       // first B elements: token ids
  float* out_filtered = out_ids + BN;        // then B*V filtered probs

  float* sMax = (float*)d_ws;
  float* Z    = sMax + BN;
  int*   thr  = (int*)(Z + BN);

  sampler_stats_kernel<<<dim3(BN), dim3(NT), 0, stream>>>(
      logits, temps, sMax, Z, thr);
  sampler_select_kernel<<<dim3(BN), dim3(NT), 0, stream>>>(
      logits, temps, topks, topps, minps, nmp, sMax, Z, thr,
      out_ids, out_filtered);
}